// VectorQuantizer_3874060501599
// MI455X (gfx1250) — compile-verified
//
#include <hip/hip_runtime.h>
#include <hip/hip_bf16.h>

typedef __attribute__((ext_vector_type(2))) float v2f;
typedef __attribute__((ext_vector_type(8))) float v8f;

#define EMB_DIM   64
#define NUM_CODES 1024
#define CB_FLOATS (NUM_CODES * EMB_DIM)      // 65536 floats = 256 KB
#define WAVES_PB  16                          // 512 threads
#define THREADS   (WAVES_PB * 32)
#define BLOCKS    256

// Dynamic LDS layout: [codebook 65536][norms 1024][wave partials 16]
#define LDS_FLOATS (CB_FLOATS + NUM_CODES + WAVES_PB)

// Async global->LDS staging (CDNA5 GLOBAL_LOAD_ASYNC_TO_LDS_B128, ASYNCcnt)
#define USE_ASYNC_STAGE 1

__device__ __forceinline__ unsigned lds_offset_of(const void* p) {
    // generic -> LDS(AS3) addrspacecast, then take the 32-bit LDS byte offset
    return (unsigned)(unsigned long long)(__attribute__((address_space(3))) const void*)p;
}

__global__ void vq_main_kernel(const float* __restrict__ x,
                               const float* __restrict__ codebook,
                               float* __restrict__ outq,
                               float* __restrict__ block_partial,
                               int ntiles) {
    extern __shared__ float smem[];
    float* cb  = smem;                 // [1024][64]
    float* nrm = smem + CB_FLOATS;     // [1024]
    float* red = nrm + NUM_CODES;      // [16]

    const int tid = threadIdx.x;

    // ---- Stage full codebook into LDS ----
#if USE_ASYNC_STAGE
    {
        const float4* src = (const float4*)codebook;
        float4*       dst = (float4*)cb;
        for (int i = tid; i < CB_FLOATS / 4; i += THREADS) {
            unsigned    lo = lds_offset_of(dst + i);
            const void* ga = (const void*)(src + i);
            asm volatile("global_load_async_to_lds_b128 %0, %1, off"
                         :: "v"(lo), "v"(ga) : "memory");
        }
        asm volatile("s_wait_asynccnt 0" ::: "memory");
    }
#else
    {
        const float4* src = (const float4*)codebook;
        float4*       dst = (float4*)cb;
        for (int i = tid; i < CB_FLOATS / 4; i += THREADS) dst[i] = src[i];
    }
#endif
    __syncthreads();

    // ---- Precompute ||e_k||^2 ----
    for (int k = tid; k < NUM_CODES; k += THREADS) {
        float s = 0.f;
        #pragma unroll 8
        for (int e = 0; e < EMB_DIM; ++e) { float v = cb[k * EMB_DIM + e]; s += v * v; }
        nrm[k] = s;
    }
    __syncthreads();

    const int wave = tid >> 5;
    const int lane = tid & 31;
    const int n    = lane & 15;   // row-in-tile (B/C/D column)
    const int hi   = lane >> 4;   // k-half selector for A/B fragments

    const int gwave  = blockIdx.x * WAVES_PB + wave;
    const int nwaves = gridDim.x * WAVES_PB;

    float lsum = 0.f;

    for (int tile = gwave; tile < ntiles; tile += nwaves) {
        const int row = tile * 16 + n;

        // ---- Prefetch next tile's X row (wave-uniform predicate; EXEC stays all-1)
        if (tile + nwaves < ntiles) {
            const char* nx = (const char*)(x + (size_t)(tile + nwaves) * 16 * EMB_DIM
                                             + (size_t)n * EMB_DIM);
            __builtin_prefetch(nx, 0, 0);
            __builtin_prefetch(nx + 128, 0, 0);
        }

        // ---- Load this lane's half of X row `row` (elements 4t+2*hi, 4t+2*hi+1) ----
        const float* xr = x + (size_t)row * EMB_DIM + 2 * hi;
        v2f xf[16];
        #pragma unroll
        for (int t = 0; t < 16; ++t) {
            float2 v = *(const float2*)(xr + 4 * t);
            xf[t].x = v.x; xf[t].y = v.y;
        }

        float bestd = 3.4028235e38f;
        int   besti = 0;

        // ---- Sweep code tiles two at a time: two independent WMMA accumulator
        //      chains to keep the XDL pipeline fed despite D->C dependencies ----
        for (int ct = 0; ct < NUM_CODES / 16; ct += 2) {
            v8f acc0 = {}, acc1 = {};
            const float* a0 = cb + ((size_t)(ct * 16 + n) * EMB_DIM + 2 * hi);
            const float* a1 = a0 + 16 * EMB_DIM;
            #pragma unroll
            for (int t = 0; t < 16; ++t) {
                float2 av0 = *(const float2*)(a0 + 4 * t);
                float2 av1 = *(const float2*)(a1 + 4 * t);
                v2f va0; va0.x = av0.x; va0.y = av0.y;
                v2f va1; va1.x = av1.x; va1.y = av1.y;
                acc0 = __builtin_amdgcn_wmma_f32_16x16x4_f32(
                    false, va0, false, xf[t], (short)0, acc0, false, false);
                acc1 = __builtin_amdgcn_wmma_f32_16x16x4_f32(
                    false, va1, false, xf[t], (short)0, acc1, false, false);
            }
            // C/D layout: lane n holds codes m=0..7 (VGPR v), lane n+16 holds m=8..15.
            // Ascending idx + strict '<' preserves lowest-index tie-break.
            #pragma unroll
            for (int v = 0; v < 8; ++v) {
                int   idx = ct * 16 + hi * 8 + v;
                float d   = nrm[idx] - 2.0f * acc0[v];   // ||x||^2 constant: argmin-invariant
                if (d < bestd) { bestd = d; besti = idx; }
            }
            #pragma unroll
            for (int v = 0; v < 8; ++v) {
                int   idx = (ct + 1) * 16 + hi * 8 + v;
                float d   = nrm[idx] - 2.0f * acc1[v];
                if (d < bestd) { bestd = d; besti = idx; }
            }
        }

        // ---- Combine lane n <-> lane n+16 (both compute identical result) ----
        {
            float od = __shfl_xor(bestd, 16, 32);
            int   oi = __shfl_xor(besti, 16, 32);
            if (od < bestd || (od == bestd && oi < besti)) { bestd = od; besti = oi; }
        }

        // ---- Write quantized row: lane pair splits the 64 floats ----
        {
            float4*       qo = (float4*)(outq + (size_t)row * EMB_DIM + hi * 32);
            const float4* qs = (const float4*)(cb + (size_t)besti * EMB_DIM + hi * 32);
            #pragma unroll
            for (int j = 0; j < 8; ++j) qo[j] = qs[j];
        }

        // ---- Loss contribution over this lane's 32 x-elements ----
        {
            const float* crow = cb + (size_t)besti * EMB_DIM + 2 * hi;
            #pragma unroll
            for (int t = 0; t < 16; ++t) {
                float dx = crow[4 * t]     - xf[t].x;
                float dy = crow[4 * t + 1] - xf[t].y;
                lsum += dx * dx + dy * dy;
            }
        }
    }

    // ---- Deterministic block reduction of loss partials ----
    #pragma unroll
    for (int off = 16; off > 0; off >>= 1) lsum += __shfl_xor(lsum, off, 32);
    if (lane == 0) red[wave] = lsum;
    __syncthreads();
    if (tid == 0) {
        float s = 0.f;
        for (int w = 0; w < WAVES_PB; ++w) s += red[w];
        block_partial[blockIdx.x] = s;
    }
}

__global__ void vq_loss_finalize(const float* __restrict__ block_partial,
                                 int nblocks, float* __restrict__ loss_out,
                                 float inv_total) {
    if (blockIdx.x == 0 && threadIdx.x == 0) {
        float s = 0.f;
        for (int i = 0; i < nblocks; ++i) s += block_partial[i];
        // loss = mean((q-x)^2) + 0.25 * mean((q-x)^2)   (stop_gradient is identity fwd)
        *loss_out = 1.25f * s * inv_total;
    }
}

extern "C" void kernel_launch(void* const* d_in, const int* in_sizes, int n_in,
                              void* d_out, int out_size, void* d_ws, size_t ws_size,
                              hipStream_t stream) {
    const float* x        = (const float*)d_in[0];   // [32,4096,64] fp32
    const float* codebook = (const float*)d_in[1];   // [1024,64] fp32
    float*       outq     = (float*)d_out;           // quantized flat + 1 loss scalar

    const int total_elems = out_size - 1;            // 32*4096*64
    const int nrows       = total_elems / EMB_DIM;   // 131072
    const int ntiles      = nrows / 16;              // 8192

    float* partial = (float*)d_ws;                   // BLOCKS floats of scratch

    const size_t shmem = (size_t)LDS_FLOATS * sizeof(float);  // ~260 KB (<320 KB/WGP)

    vq_main_kernel<<<BLOCKS, THREADS, shmem, stream>>>(x, codebook, outq, partial, ntiles);
    vq_loss_finalize<<<1, 32, 0, stream>>>(partial, BLOCKS, outq + total_elems,
                                           1.0f / (float)total_elems);
}